// Eq1to2_58815282152310
// MI455X (gfx1250) — compile-verified
//
#include <hip/hip_runtime.h>
#include <hip/hip_bf16.h>
#include <math.h>

typedef float v2f __attribute__((ext_vector_type(2)));
typedef float v8f __attribute__((ext_vector_type(8)));

#define B_ 8
#define N_ 128
#define C_ 64
#define S_ 64
#define BT_ 20   // basis_total

__device__ __forceinline__ float lrelu(float v) { return v >= 0.0f ? v : 0.01f * v; }

// ---------------------------------------------------------------------------
// K0: per-(n,d) aggregations over i: sum/49, sum/nobj, max, min
// agg layout: agg[n*4*C + a*C + d]
// ---------------------------------------------------------------------------
__global__ void k_agg(const float* __restrict__ x, const float* __restrict__ nobj,
                      float* __restrict__ agg) {
    int n = blockIdx.x;
    int d = threadIdx.x;
    const float* xp = x + (size_t)n * N_ * C_ + d;
    float s = 0.0f, mx = -INFINITY, mn = INFINITY;
#pragma unroll 4
    for (int i = 0; i < N_; ++i) {
        float v = xp[i * C_];
        s += v;
        mx = fmaxf(mx, v);
        mn = fminf(mn, v);
    }
    float* ap = agg + n * 4 * C_;
    ap[0 * C_ + d] = s * (1.0f / 49.0f);
    ap[1 * C_ + d] = s / nobj[n];
    ap[2 * C_ + d] = mx;
    ap[3 * C_ + d] = mn;
}

// ---------------------------------------------------------------------------
// K1: fold coefs over the 4 aggregation configs for basis slots 0/1/2
// wf[m*4096 + d*64 + s] = sum_a coefs[d*1280 + s*20 + 5a + m]
//   m=0 -> diag x_i term (R),  m=1 -> x_j term (Q),  m=2 -> x_i term (P)
// ---------------------------------------------------------------------------
__global__ void k_fold(const float* __restrict__ coefs, float* __restrict__ wf) {
    int idx = blockIdx.x * blockDim.x + threadIdx.x;   // 3*64*64 = 12288
    int m = idx >> 12;
    int d = (idx >> 6) & 63;
    int s = idx & 63;
    const float* cp = coefs + d * (S_ * BT_) + s * BT_ + m;
    wf[idx] = cp[0] + cp[5] + cp[10] + cp[15];
}

// ---------------------------------------------------------------------------
// K2: batch constants  Eb[n,s] = sum_a agg_a . coefs[:,s,5a+4] + bias[s]
//                       F [n,s] = sum_a agg_a . coefs[:,s,5a+3]
// ---------------------------------------------------------------------------
__global__ void k_ef(const float* __restrict__ agg, const float* __restrict__ coefs,
                     const float* __restrict__ bias,
                     float* __restrict__ Eb, float* __restrict__ F) {
    int n = blockIdx.x;
    int s = threadIdx.x;
    float e = 0.0f, f = 0.0f;
#pragma unroll
    for (int a = 0; a < 4; ++a) {
        const float* ag = agg + n * 4 * C_ + a * C_;
        for (int d = 0; d < C_; ++d) {
            const float* cp = coefs + d * (S_ * BT_) + s * BT_ + 5 * a;
            f += ag[d] * cp[3];
            e += ag[d] * cp[4];
        }
    }
    Eb[n * S_ + s] = e + bias[s];
    F[n * S_ + s]  = f;
}

// ---------------------------------------------------------------------------
// K3: WMMA GEMM  pqr[m][row][s] = sum_d x[row,d] * wf[m][d][s]
// X is [1024 x 64] (rows = n*128+i), wf is 3 matrices of [64 x 64].
// One block = 4 waves; wave w owns S-tile [16w,16w+16) for all three m.
// V_WMMA_F32_16X16X4_F32 fragment layout (ISA 7.12.2):
//   A 16x4: lane%16 = M, VGPR v in {0,1}: K = (lane/16)*2 + v
//   B 4x16: lane%16 = N, VGPR v in {0,1}: K = (lane/16)*2 + v
//   C/D:    lane%16 = N, VGPR r: M = r + (lane/16)*8
// ---------------------------------------------------------------------------
__global__ void k_gemm(const float* __restrict__ x, const float* __restrict__ wf,
                       float* __restrict__ pqr) {
    int wave = threadIdx.x >> 5;       // 0..3 -> S col tile
    int lane = threadIdx.x & 31;
    int rl   = lane & 15;
    int kh   = lane >> 4;              // 0 or 1
    int row  = blockIdx.x * 16 + rl;   // A-fragment row
    int col  = wave * 16 + rl;         // B/D-fragment column

    const float* w0 = wf;              // m=0 (R)
    const float* w1 = wf + 4096;       // m=1 (Q)
    const float* w2 = wf + 8192;       // m=2 (P)

    v8f acc0 = {}, acc1 = {}, acc2 = {};

#pragma unroll 4
    for (int k0 = 0; k0 < C_; k0 += 4) {
        int ka = k0 + kh * 2;
        v2f a;
        a.x = x[row * C_ + ka + 0];
        a.y = x[row * C_ + ka + 1];
        v2f b0, b1, b2;
        b0.x = w0[(ka + 0) * S_ + col];  b0.y = w0[(ka + 1) * S_ + col];
        b1.x = w1[(ka + 0) * S_ + col];  b1.y = w1[(ka + 1) * S_ + col];
        b2.x = w2[(ka + 0) * S_ + col];  b2.y = w2[(ka + 1) * S_ + col];
        acc0 = __builtin_amdgcn_wmma_f32_16x16x4_f32(false, a, false, b0, (short)0, acc0, false, false);
        acc1 = __builtin_amdgcn_wmma_f32_16x16x4_f32(false, a, false, b1, (short)0, acc1, false, false);
        acc2 = __builtin_amdgcn_wmma_f32_16x16x4_f32(false, a, false, b2, (short)0, acc2, false, false);
    }

#pragma unroll
    for (int r = 0; r < 8; ++r) {
        int mrow = blockIdx.x * 16 + r + kh * 8;
        pqr[0 * 65536 + mrow * S_ + col] = acc0[r];
        pqr[1 * 65536 + mrow * S_ + col] = acc1[r];
        pqr[2 * 65536 + mrow * S_ + col] = acc2[r];
    }
}

// ---------------------------------------------------------------------------
// K4: streaming output.  One block per (n,i) row of the NxN grid.
// out[n,i,j,s] = lrelu(P[n,i,s]+Eb[n,s] + Q[n,j,s] + δ_ij(R[n,i,s]+F[n,s])) * mask
// Thread t: s-group = t&15 (float4 of s), j-lane = t>>4; each wave stores
// 512 contiguous bytes per iteration.
// ---------------------------------------------------------------------------
__global__ void k_out(const float* __restrict__ pqr, const float* __restrict__ Eb,
                      const float* __restrict__ F, const float* __restrict__ mask,
                      float* __restrict__ out) {
    int bx = blockIdx.x;            // n*128 + i
    int n  = bx >> 7;
    int i  = bx & 127;
    int t  = threadIdx.x;
    int sg = t & 15;                // which float4 of S
    int jl = t >> 4;                // 0..15

    const float4* R4 = (const float4*)(pqr);              // m=0
    const float4* Q4 = (const float4*)(pqr + 65536);      // m=1
    const float4* P4 = (const float4*)(pqr + 131072);     // m=2
    const float4* E4 = (const float4*)Eb;
    const float4* F4 = (const float4*)F;

    float4 p = P4[bx * 16 + sg];
    float4 e = E4[n * 16 + sg];
    p.x += e.x; p.y += e.y; p.z += e.z; p.w += e.w;

    float4 r = R4[bx * 16 + sg];
    float4 f = F4[n * 16 + sg];
    r.x += f.x; r.y += f.y; r.z += f.z; r.w += f.w;

    const float* mrow = mask + ((size_t)n * N_ + i) * N_;
    float4* O4 = (float4*)out;

    for (int j = jl; j < N_; j += 16) {
        float4 q = Q4[(n * N_ + j) * 16 + sg];
        float4 v;
        v.x = p.x + q.x; v.y = p.y + q.y; v.z = p.z + q.z; v.w = p.w + q.w;
        if (j == i) { v.x += r.x; v.y += r.y; v.z += r.z; v.w += r.w; }
        v.x = lrelu(v.x); v.y = lrelu(v.y); v.z = lrelu(v.z); v.w = lrelu(v.w);
        float m = mrow[j];
        v.x *= m; v.y *= m; v.z *= m; v.w *= m;
        O4[((size_t)bx * N_ + j) * 16 + sg] = v;
    }
}

// ---------------------------------------------------------------------------
extern "C" void kernel_launch(void* const* d_in, const int* in_sizes, int n_in,
                              void* d_out, int out_size, void* d_ws, size_t ws_size,
                              hipStream_t stream) {
    const float* x     = (const float*)d_in[0];   // [8,128,64]
    const float* mask  = (const float*)d_in[1];   // [8,128,128,1]
    const float* nobj  = (const float*)d_in[2];   // [8]
    const float* coefs = (const float*)d_in[3];   // [64,64,20]
    const float* bias  = (const float*)d_in[4];   // [64]
    float* out = (float*)d_out;                   // [8,128,128,64]

    float* ws  = (float*)d_ws;
    float* agg = ws;              // 8*4*64      = 2048
    float* wf  = ws + 2048;       // 3*64*64     = 12288
    float* Eb  = ws + 14336;      // 8*64        = 512
    float* F   = ws + 14848;      // 8*64        = 512
    float* pqr = ws + 15360;      // 3*1024*64   = 196608   (~830 KB total)

    k_agg <<<B_, C_, 0, stream>>>(x, nobj, agg);
    k_fold<<<48, 256, 0, stream>>>(coefs, wf);
    k_ef  <<<B_, S_, 0, stream>>>(agg, coefs, bias, Eb, F);
    k_gemm<<<(B_ * N_) / 16, 128, 0, stream>>>(x, wf, pqr);
    k_out <<<B_ * N_, 256, 0, stream>>>(pqr, Eb, F, mask, out);
}